// GNNModel_18193481466190
// MI455X (gfx1250) — compile-verified
//
#include <hip/hip_runtime.h>

typedef __attribute__((ext_vector_type(2))) float v2f;
typedef __attribute__((ext_vector_type(8))) float v8f;

constexpr int NNODES = 50000;
constexpr int NEDGES = 800000;
constexpr int INF    = 256;
constexpr int HID    = 256;
constexpr int NCLS   = 64;

// ---------------------------------------------------------------- utilities
__global__ void zero_f32(float* __restrict__ p, int n) {
    int i = blockIdx.x * blockDim.x + threadIdx.x;
    if (i < n) p[i] = 0.0f;
}

__global__ void degree_kernel(const int* __restrict__ src, const int* __restrict__ dst,
                              float* __restrict__ dego, float* __restrict__ degi, int E) {
    int e = blockIdx.x * blockDim.x + threadIdx.x;
    if (e < E) {
        atomicAdd(&dego[src[e]], 1.0f);
        atomicAdd(&degi[dst[e]], 1.0f);
    }
}

__global__ void invsqrt_kernel(float* __restrict__ p, int n) {
    int i = blockIdx.x * blockDim.x + threadIdx.x;
    if (i < n) p[i] = rsqrtf(fmaxf(p[i], 1.0f));
}

__global__ void edgenorm_kernel(const int* __restrict__ src, const int* __restrict__ dst,
                                const float* __restrict__ iso, const float* __restrict__ isi,
                                float* __restrict__ enorm, int E) {
    int e = blockIdx.x * blockDim.x + threadIdx.x;
    if (e < E) enorm[e] = iso[src[e]] * isi[dst[e]];
}

// out[i,j] = max(out[i,j] + b[j & mask], 0)   (relu == true)  or  += b (relu == false)
template <bool RELU>
__global__ void bias_kernel(float* __restrict__ x, const float* __restrict__ b,
                            int total, int mask) {
    int i = blockIdx.x * blockDim.x + threadIdx.x;
    if (i < total) {
        float v = x[i] + b[i & mask];
        x[i] = RELU ? fmaxf(v, 0.0f) : v;
    }
}

// ------------------------------------------------------- fp32 WMMA GEMM
// C[M x N] = A[M x K] * B[K x N], row-major, M = 16*gridDim.x, N/K multiples of 16.
// One wave computes one 16x16 tile of C; K consumed 4 at a time by
// v_wmma_f32_16x16x4_f32. tileN = blockIdx.y * wavesPerBlock + waveId.
__global__ void wmma_gemm_f32(const float* __restrict__ A, const float* __restrict__ B,
                              float* __restrict__ C, int K, int N) {
    const int lane  = threadIdx.x & 31;
    const int wave  = threadIdx.x >> 5;
    const int waves = blockDim.x >> 5;
    const int half  = lane >> 4;      // 0: lanes 0-15, 1: lanes 16-31
    const int l16   = lane & 15;

    const int row0 = blockIdx.x * 16;
    const int col0 = (blockIdx.y * waves + wave) * 16;
    if (col0 >= N) return;            // wave-uniform: EXEC stays all-ones

    // A-frag: lane holds A[row0+l16][k0 + 2*half + {0,1}]   (ISA 16x4 layout)
    const float* Ap = A + (long)(row0 + l16) * K + 2 * half;
    // B-frag: lane holds B[k0 + 2*half + {0,1}][col0+l16]
    const float* Bp = B + (long)(2 * half) * N + col0 + l16;

    v8f acc = {};
#pragma unroll 8
    for (int k0 = 0; k0 < K; k0 += 4) {
        v2f a = *reinterpret_cast<const v2f*>(Ap + k0);   // 8B-aligned (K,k0 even)
        v2f b;
        b.x = Bp[(long)k0 * N];
        b.y = Bp[(long)(k0 + 1) * N];
        acc = __builtin_amdgcn_wmma_f32_16x16x4_f32(false, a, false, b,
                                                    (short)0, acc, false, false);
    }

    // D layout: VGPR r -> row = row0 + r + 8*half, col = col0 + l16
    float* Cp = C + (long)(row0 + 8 * half) * N + col0 + l16;
#pragma unroll
    for (int r = 0; r < 8; ++r) Cp[(long)r * N] = acc[r];
}

// ------------------------------------------------------- edge scatter-add
// AGG[dst[e], :] += enorm[e] * H[src[e], :]
// F = 1<<LOGF features; each thread: one edge, 4 consecutive feats.
template <int LOGF>
__global__ void scatter_kernel(const float* __restrict__ H, const int* __restrict__ src,
                               const int* __restrict__ dst, const float* __restrict__ enorm,
                               float* __restrict__ AGG, int E) {
    constexpr int F    = 1 << LOGF;
    constexpr int LOGC = LOGF - 2;          // float4 chunks per edge
    int idx = blockIdx.x * blockDim.x + threadIdx.x;
    int e = idx >> LOGC;
    if (e >= E) return;
    int c = (idx & ((1 << LOGC) - 1)) << 2;
    int s = src[e], d = dst[e];
    float w = enorm[e];
    float4 h = *reinterpret_cast<const float4*>(H + (long)s * F + c);
    float* o = AGG + (long)d * F + c;
    atomicAdd(o + 0, h.x * w);
    atomicAdd(o + 1, h.y * w);
    atomicAdd(o + 2, h.z * w);
    atomicAdd(o + 3, h.w * w);
}

// ---------------------------------------------------------------- launcher
extern "C" void kernel_launch(void* const* d_in, const int* in_sizes, int n_in,
                              void* d_out, int out_size, void* d_ws, size_t ws_size,
                              hipStream_t stream) {
    const float* X  = (const float*)d_in[0];
    const int*   ei = (const int*)d_in[1];
    const float* W1 = (const float*)d_in[2];
    const float* b1 = (const float*)d_in[3];
    const float* W2 = (const float*)d_in[4];
    const float* b2 = (const float*)d_in[5];
    float* out = (float*)d_out;

    const int* src = ei;
    const int* dst = ei + NEDGES;

    float* f     = (float*)d_ws;
    float* H1    = f;                               // 50000*256 = 12.8M floats
    float* AGG   = f + (long)NNODES * HID;          // 12.8M floats
    float* dego  = AGG + (long)NNODES * HID;        // 50000
    float* degi  = dego + NNODES;                   // 50000
    float* enorm = degi + NNODES;                   // 800000
    float* H2    = H1;                              // reuse H1 region for layer-2 GEMM out

    const int T = 256;
    auto blks = [](long n, int t) { return (int)((n + t - 1) / t); };

    // 1) symmetric degree normalization
    zero_f32<<<blks(2 * NNODES, T), T, 0, stream>>>(dego, 2 * NNODES);
    degree_kernel<<<blks(NEDGES, T), T, 0, stream>>>(src, dst, dego, degi, NEDGES);
    invsqrt_kernel<<<blks(2 * NNODES, T), T, 0, stream>>>(dego, 2 * NNODES);
    edgenorm_kernel<<<blks(NEDGES, T), T, 0, stream>>>(src, dst, dego, degi, enorm, NEDGES);

    // 2) H1 = X @ W1   (fp32 WMMA: 3125 x 16 tiles of 16x16)
    wmma_gemm_f32<<<dim3(NNODES / 16, HID / 16 / 8), 256, 0, stream>>>(X, W1, H1, INF, HID);

    // 3) AGG = scatter(enorm * H1[src] -> dst); h = relu(AGG + b1) in-place
    zero_f32<<<blks((long)NNODES * HID, T), T, 0, stream>>>(AGG, NNODES * HID);
    scatter_kernel<8><<<blks((long)NEDGES * (HID / 4), T), T, 0, stream>>>(
        H1, src, dst, enorm, AGG, NEDGES);
    bias_kernel<true><<<blks((long)NNODES * HID, T), T, 0, stream>>>(
        AGG, b1, NNODES * HID, HID - 1);

    // 4) H2 = h @ W2   (N=64 -> 4 tiles, 4 waves per block)
    wmma_gemm_f32<<<dim3(NNODES / 16, 1), 128, 0, stream>>>(AGG, W2, H2, HID, NCLS);

    // 5) out = scatter(enorm * H2[src] -> dst) + b2
    zero_f32<<<blks((long)NNODES * NCLS, T), T, 0, stream>>>(out, NNODES * NCLS);
    scatter_kernel<6><<<blks((long)NEDGES * (NCLS / 4), T), T, 0, stream>>>(
        H2, src, dst, enorm, out, NEDGES);
    bias_kernel<false><<<blks((long)NNODES * NCLS, T), T, 0, stream>>>(
        out, b2, NNODES * NCLS, NCLS - 1);
}